// Bitonic_70025146794247
// MI455X (gfx1250) — compile-verified
//
#include <hip/hip_runtime.h>
#include <stdint.h>

// Bitonic sort of 32768 rows x 1024 f32, ascending.
// One wave32 per row; lane l holds elements [l*32, l*32+32) in 32 VGPRs.
// Strides 1..16: in-register min/max. Strides 32..512: shfl_xor across lanes.
// Lane-dependent sort directions (stages k>=32) are handled by sign-flipping
// values in descending lanes at stage boundaries, so every compare-exchange
// is a plain ascending min/max (VOPD-packable, no source modifiers).
// Row load: TDM tensor_load_to_lds with LDS padding (16B per 128B) so lane
// chunk reads have a 144B stride (2-way LDS bank conflicts instead of 16-way).

#define ROW_N 1024
#define WAVES_PER_BLOCK 8
#define LDS_ROW_BYTES 4608  // 4096 + 16B pad per 128B (32 pads * 16B)

typedef uint32_t v4u __attribute__((ext_vector_type(4)));
typedef int32_t  v8i __attribute__((ext_vector_type(8)));
typedef int32_t  v4i __attribute__((ext_vector_type(4)));
typedef float    vf4 __attribute__((ext_vector_type(4)));

#if defined(__gfx1250__) && __has_builtin(__builtin_amdgcn_tensor_load_to_lds)
#define USE_TDM 1
#else
#define USE_TDM 0
#endif

__device__ __forceinline__ void wait_tensorcnt0() {
#if __has_builtin(__builtin_amdgcn_s_wait_tensorcnt)
    __builtin_amdgcn_s_wait_tensorcnt(0);
#else
    asm volatile("s_wait_tensorcnt 0x0" ::: "memory");
#endif
    asm volatile("" ::: "memory");  // compiler-level fence: LDS now holds TDM data
}

__global__ __launch_bounds__(WAVES_PER_BLOCK * 32)
void Bitonic_sort_kernel(const float* __restrict__ x, float* __restrict__ out, int n_rows) {
#if USE_TDM
    __shared__ __align__(128) char smem[WAVES_PER_BLOCK * LDS_ROW_BYTES];
#endif
    const int lane   = (int)(threadIdx.x & 31u);
    const int waveId = __builtin_amdgcn_readfirstlane((int)(threadIdx.x >> 5));
    const int row    = (int)blockIdx.x * WAVES_PER_BLOCK + waveId;
    if (row >= n_rows) return;

    float v[32];

#if USE_TDM
    {
        // LDS byte offset of this wave's row buffer: generic shared pointer's
        // low 32 bits are the workgroup-relative LDS address (flat aperture).
        const uint32_t lds_addr =
            (uint32_t)(uintptr_t)smem + (uint32_t)waveId * LDS_ROW_BYTES;
        const uint64_t ga = (uint64_t)(uintptr_t)(x + (size_t)row * ROW_N);

        // ---- D# group 0: count=1, lds_addr, global_addr, type=2 ----
        v4u g0;
        g0.x = 1u;                                           // count=1, user mode
        g0.y = lds_addr;                                     // bits [63:32]
        g0.z = (uint32_t)(ga & 0xFFFFFFFFu);                 // global_addr[31:0]
        g0.w = (uint32_t)((ga >> 32) & 0x01FFFFFFu)          // global_addr[56:32]
             | (2u << 30);                                   // type=2 ("image")
        // ---- D# group 1 ----
        // data_size=2 (4B), pad_enable, pad_interval=4 (32 DWORDs = 128B),
        // pad_amount=3 (4 DWORDs = 16B). 1D row: dims in 4B elements.
        const uint32_t tdim0 = ROW_N, tdim1 = 1, tile0 = ROW_N, tile1 = 1, tile2 = 0;
        const uint32_t s0 = ROW_N, s1 = ROW_N;               // strides, 48-bit (high=0)
        v8i g1;
        g1[0] = (int32_t)((2u << 16) | (1u << 20) | (4u << 22) | (3u << 25));
        g1[1] = (int32_t)((tdim0 & 0xFFFFu) << 16);          // abar_addr=0 | tdim0.lo
        g1[2] = (int32_t)((tdim0 >> 16) | ((tdim1 & 0xFFFFu) << 16));
        g1[3] = (int32_t)((tdim1 >> 16) | (tile0 << 16));
        g1[4] = (int32_t)(tile1 | (tile2 << 16));
        g1[5] = (int32_t)s0;                                 // stride0[31:0]
        g1[6] = (int32_t)((s1 & 0xFFFFu) << 16);             // stride0[47:32]=0 | stride1.lo
        g1[7] = (int32_t)(s1 >> 16);                         // stride1[47:16]
        v4i gz4 = {0, 0, 0, 0};
        v8i gz8 = {0, 0, 0, 0, 0, 0, 0, 0};

        __builtin_amdgcn_tensor_load_to_lds(g0, g1, gz4, gz4, gz8, 0);
        wait_tensorcnt0();

        // Lane chunk sits at padded stride 144B: elem i -> 4*i + 16*(i/32).
        const char* lanep = smem + (size_t)waveId * LDS_ROW_BYTES + (size_t)lane * 144;
#pragma unroll
        for (int q = 0; q < 8; ++q) {
            vf4 t = *(const vf4*)(lanep + q * 16);
            v[4*q+0] = t.x; v[4*q+1] = t.y; v[4*q+2] = t.z; v[4*q+3] = t.w;
        }
    }
#else
    {
        const vf4* src = (const vf4*)(x + (size_t)row * ROW_N + (size_t)lane * 32);
#pragma unroll
        for (int q = 0; q < 8; ++q) {
            vf4 t = __builtin_nontemporal_load(src + q);
            v[4*q+0] = t.x; v[4*q+1] = t.y; v[4*q+2] = t.z; v[4*q+3] = t.w;
        }
    }
#endif

    // ---- Bitonic network: element index i = lane*32 + r, ascending overall.
    // Stage (k,j): compare i with i^j; original direction: asc iff (i&k)==0.
    // For k>=32 the direction bit lives in the lane bits; we sign-flip values
    // in descending lanes at stage entry (merging with the previous stage's
    // un-flip), so all compare-exchanges below are plain ascending.
#pragma unroll
    for (int k = 2; k <= ROW_N; k <<= 1) {
        if (k >= 32) {
            // Enter stage-k flip state: xor sign with F_{k/2} ^ F_k.
            const int bprev = (k == 32) ? 0 : ((k >> 1) >> 5);
            const int bcur  = (k >> 5) & 31;   // k=1024 -> 0 (always ascending)
            const uint32_t fprev = (lane & bprev) ? 0x80000000u : 0u;
            const uint32_t fcur  = (lane & bcur)  ? 0x80000000u : 0u;
            const uint32_t fl = fprev ^ fcur;
#pragma unroll
            for (int r = 0; r < 32; ++r)
                v[r] = __uint_as_float(__float_as_uint(v[r]) ^ fl);
        }
#pragma unroll
        for (int j = k >> 1; j > 0; j >>= 1) {
            if (j >= 32) {
                // Cross-lane butterfly: partner lane = lane ^ (j>>5).
                const int  m       = j >> 5;
                const bool keepmin = ((lane & m) == 0);
                float w[32];
#pragma unroll
                for (int r = 0; r < 32; ++r) w[r] = __shfl_xor(v[r], m, 32);
#pragma unroll
                for (int r = 0; r < 32; ++r) {
                    float mn = fminf(v[r], w[r]);
                    float mx = fmaxf(v[r], w[r]);
                    v[r] = keepmin ? mn : mx;
                }
            } else if (k <= 16) {
                // In-register, compile-time direction per register pair.
#pragma unroll
                for (int r = 0; r < 32; ++r) {
                    if (r & j) continue;
                    const int  r2  = r | j;
                    const bool asc = ((r & k) == 0);
                    float a = v[r], b = v[r2];
                    float mn = fminf(a, b), mx = fmaxf(a, b);
                    v[r]  = asc ? mn : mx;
                    v[r2] = asc ? mx : mn;
                }
            } else {
                // In-register, ascending (direction folded into sign flip).
#pragma unroll
                for (int r = 0; r < 32; ++r) {
                    if (r & j) continue;
                    const int r2 = r | j;
                    float a = v[r], b = v[r2];
                    v[r]  = fminf(a, b);
                    v[r2] = fmaxf(a, b);
                }
            }
        }
    }

    // ---- Store sorted chunk; streaming (non-temporal) b128 stores.
    float* drow = out + (size_t)row * ROW_N + (size_t)lane * 32;
#pragma unroll
    for (int q = 0; q < 8; ++q) {
        vf4 t;
        t.x = v[4*q+0]; t.y = v[4*q+1]; t.z = v[4*q+2]; t.w = v[4*q+3];
        __builtin_nontemporal_store(t, (vf4*)(drow + 4*q));
    }
}

extern "C" void kernel_launch(void* const* d_in, const int* in_sizes, int n_in,
                              void* d_out, int out_size, void* d_ws, size_t ws_size,
                              hipStream_t stream) {
    const float* x  = (const float*)d_in[0];
    float* out      = (float*)d_out;
    const int n_rows = in_sizes[0] / ROW_N;                  // 16*64*32 = 32768
    const int blocks = (n_rows + WAVES_PER_BLOCK - 1) / WAVES_PER_BLOCK;
    Bitonic_sort_kernel<<<blocks, WAVES_PER_BLOCK * 32, 0, stream>>>(x, out, n_rows);
}